// MPAttn_4363686772830
// MI455X (gfx1250) — compile-verified
//
#include <hip/hip_runtime.h>

#define DIMH 128
#define NHEADS 8

typedef __attribute__((ext_vector_type(2))) float v2f;
typedef __attribute__((ext_vector_type(8))) float v8f;

// ---------------------------------------------------------------------------
// Zero-init: h output region and segmax/segsum scratch (contiguous block b).
// ---------------------------------------------------------------------------
__global__ __launch_bounds__(256) void init_zero(float* __restrict__ a, long na,
                                                 float* __restrict__ b, long nb) {
    long i = (long)blockIdx.x * blockDim.x + threadIdx.x;
    long stride = (long)gridDim.x * blockDim.x;
    for (long k = i; k < na; k += stride) a[k] = 0.0f;
    for (long k = i; k < nb; k += stride) b[k] = 0.0f;
}

// ---------------------------------------------------------------------------
// Q/K/V projection: out = x @ W^T + b using V_WMMA_F32_16X16X4_F32.
// One wave32 computes one 16x16 tile of one of {Q,K,V} (blockIdx.z selects).
// A (16x4 f32) layout: lanes 0-15 hold rows M=0..15, VGPR0={K0|K2}, VGPR1={K1|K3}
// B (4x16 f32) layout: VGPR0 = {K0 row | K2 row}, VGPR1 = {K1 row | K3 row}
// C/D (16x16 f32): VGPR r: lanes0-15 -> M=r, N=lane; lanes16-31 -> M=8+r.
// ---------------------------------------------------------------------------
__global__ __launch_bounds__(32) void qkv_proj_wmma(
    const float* __restrict__ x,
    const float* __restrict__ WQ, const float* __restrict__ bQ,
    const float* __restrict__ WK, const float* __restrict__ bK,
    const float* __restrict__ WV, const float* __restrict__ bV,
    float* __restrict__ Q, float* __restrict__ Kd, float* __restrict__ Vd,
    int n)
{
    const float* W; const float* bias; float* out;
    if (blockIdx.z == 0)      { W = WQ; bias = bQ; out = Q;  }
    else if (blockIdx.z == 1) { W = WK; bias = bK; out = Kd; }
    else                      { W = WV; bias = bV; out = Vd; }

    const int lane = threadIdx.x;      // 0..31
    const int half = lane >> 4;        // 0: K pair {0,1}; 1: K pair {2,3}
    const int l    = lane & 15;
    const int m0   = blockIdx.x * 16;  // node-row tile
    const int n0   = blockIdx.y * 16;  // output-column tile

    int arow = m0 + l; if (arow >= n) arow = n - 1;     // clamp (EXEC must stay full)
    const float* xr = x + (long)arow * DIMH + half * 2; // A source row
    const float* wr = W + (long)(n0 + l) * DIMH + half * 2; // W row n0+l = B column

    // C init with bias: C[m][n] = bias[n] for every row -> same value in all 8 VGPRs
    float bv = bias[n0 + l];
    v8f acc;
#pragma unroll
    for (int r = 0; r < 8; ++r) acc[r] = bv;

#pragma unroll
    for (int kk = 0; kk < DIMH; kk += 4) {
        v2f a; a.x = xr[kk]; a.y = xr[kk + 1];   // A[m][kk+2*half .. +1]
        v2f b; b.x = wr[kk]; b.y = wr[kk + 1];   // B[kk+2*half..][n] = W[n][k]
        acc = __builtin_amdgcn_wmma_f32_16x16x4_f32(
            /*neg_a=*/false, a, /*neg_b=*/false, b,
            /*c_mod=*/(short)0, acc, /*reuse_a=*/false, /*reuse_b=*/false);
    }

#pragma unroll
    for (int r = 0; r < 8; ++r) {
        int row = m0 + half * 8 + r;
        if (row < n) out[(long)row * DIMH + n0 + l] = acc[r];
    }
}

// ---------------------------------------------------------------------------
// Pass 2: per-edge scores. One wave per edge; lane -> float4 chunk of the
// 128-float Q[dst]/K[src] rows (coalesced 512B). 4 lanes per head, shuffle
// reduce, add edge bias, store score, atomic segment-max via monotonic key.
// ---------------------------------------------------------------------------
__global__ __launch_bounds__(256) void edge_scores_kernel(
    const float* __restrict__ Q, const float* __restrict__ K,
    const int* __restrict__ ei, const float* __restrict__ ebias,
    float* __restrict__ scores, unsigned* __restrict__ segmax, int E)
{
    int gtid  = blockIdx.x * blockDim.x + threadIdx.x;
    int wave  = gtid >> 5;
    int lane  = threadIdx.x & 31;
    int nwv   = (gridDim.x * blockDim.x) >> 5;
    for (int e = wave; e < E; e += nwv) {
        int src = ei[e];
        int dst = ei[E + e];
        float4 q = ((const float4*)(Q + (long)dst * DIMH))[lane];
        float4 k = ((const float4*)(K + (long)src * DIMH))[lane];
        float p = q.x * k.x + q.y * k.y + q.z * k.z + q.w * k.w;
        p += __shfl_xor(p, 1, 32);
        p += __shfl_xor(p, 2, 32);
        if ((lane & 3) == 0) {
            int hh = lane >> 2;
            float s = p * 0.25f + ebias[(long)e * NHEADS + hh]; // 1/sqrt(16)
            scores[(long)e * NHEADS + hh] = s;
            unsigned bits = __float_as_uint(s);
            unsigned key  = (bits & 0x80000000u) ? ~bits : (bits | 0x80000000u);
            atomicMax(segmax + (long)dst * NHEADS + hh, key);
        }
    }
}

// ---------------------------------------------------------------------------
// Pass 3: ex = exp(score - segmax[dst]); store in place; segsum += ex.
// ---------------------------------------------------------------------------
__global__ __launch_bounds__(256) void edge_exp_kernel(
    const int* __restrict__ ei, float* __restrict__ scores,
    const unsigned* __restrict__ segmax, float* __restrict__ segsum, int E)
{
    long i      = (long)blockIdx.x * blockDim.x + threadIdx.x;
    long total  = (long)E * NHEADS;
    long stride = (long)gridDim.x * blockDim.x;
    for (; i < total; i += stride) {
        int e  = (int)(i >> 3);
        int hh = (int)(i & 7);
        int dst = ei[E + e];
        unsigned key  = segmax[(long)dst * NHEADS + hh];
        unsigned bits = (key & 0x80000000u) ? (key & 0x7FFFFFFFu) : ~key;
        float mx = __uint_as_float(bits);
        float ex = __expf(scores[i] - mx);
        scores[i] = ex;
        atomicAdd(segsum + (long)dst * NHEADS + hh, ex);
    }
}

// ---------------------------------------------------------------------------
// Pass 4: alpha = ex / segsum[dst]; h[dst] += alpha * V[src].
// Wave per edge; lane -> float4 chunk; 4 float atomic adds per lane (h is
// L2-resident, 25.6MB << 192MB L2).
// ---------------------------------------------------------------------------
__global__ __launch_bounds__(256) void edge_aggregate_kernel(
    const float* __restrict__ V, const int* __restrict__ ei,
    float* __restrict__ alpha, const float* __restrict__ segsum,
    float* __restrict__ h, int E)
{
    int gtid = blockIdx.x * blockDim.x + threadIdx.x;
    int wave = gtid >> 5;
    int lane = threadIdx.x & 31;
    int nwv  = (gridDim.x * blockDim.x) >> 5;
    for (int e = wave; e < E; e += nwv) {
        int src = ei[e];
        int dst = ei[E + e];
        int hh  = lane >> 2;
        float ex = alpha[(long)e * NHEADS + hh];        // read-before-write, wave-local
        float ss = segsum[(long)dst * NHEADS + hh];
        float a  = ex / ss;
        if ((lane & 3) == 0) alpha[(long)e * NHEADS + hh] = a;
        float4 v = ((const float4*)(V + (long)src * DIMH))[lane];
        float* hb = h + (long)dst * DIMH + lane * 4;
        atomicAdd(hb + 0, a * v.x);
        atomicAdd(hb + 1, a * v.y);
        atomicAdd(hb + 2, a * v.z);
        atomicAdd(hb + 3, a * v.w);
    }
}

// ---------------------------------------------------------------------------
extern "C" void kernel_launch(void* const* d_in, const int* in_sizes, int n_in,
                              void* d_out, int out_size, void* d_ws, size_t ws_size,
                              hipStream_t stream) {
    const float* x     = (const float*)d_in[0];
    const int*   ei    = (const int*)  d_in[1];
    const float* ebias = (const float*)d_in[2];
    const float* WQ    = (const float*)d_in[3];
    const float* bQ    = (const float*)d_in[4];
    const float* WK    = (const float*)d_in[5];
    const float* bK    = (const float*)d_in[6];
    const float* WV    = (const float*)d_in[7];
    const float* bV    = (const float*)d_in[8];

    const int N = in_sizes[0] / DIMH;
    const int E = in_sizes[1] / 2;

    float* out   = (float*)d_out;
    float* h     = out;                          // [N,128]
    float* alpha = out + (long)N * DIMH;         // [E,8]  (scores -> ex -> alpha)

    float* Q = (float*)d_ws;                     // [N,128]
    float* K = Q + (long)N * DIMH;               // [N,128]
    float* V = K + (long)N * DIMH;               // [N,128]
    unsigned* segmax = (unsigned*)(V + (long)N * DIMH);   // [N,8] monotone keys
    float*    segsum = (float*)(segmax + (long)N * NHEADS); // [N,8]

    // 1) zero h and (segmax|segsum) — contiguous N*16 words
    init_zero<<<2048, 256, 0, stream>>>(h, (long)N * DIMH,
                                        (float*)segmax, (long)N * NHEADS * 2);

    // 2) Q/K/V projections via f32 WMMA
    dim3 pg((N + 15) / 16, DIMH / 16, 3);
    qkv_proj_wmma<<<pg, 32, 0, stream>>>(x, WQ, bQ, WK, bK, WV, bV, Q, K, V, N);

    // 3) per-edge scores + segment max
    int nblk_edge = (E + 7) / 8;                 // 8 waves per 256-thread block
    edge_scores_kernel<<<nblk_edge, 256, 0, stream>>>(Q, K, ei, ebias, alpha, segmax, E);

    // 4) exp + segment sum
    long totEH = (long)E * NHEADS;
    int nblk_eh = (int)((totEH + 255) / 256);
    edge_exp_kernel<<<nblk_eh, 256, 0, stream>>>(ei, alpha, segmax, segsum, E);

    // 5) normalize + weighted scatter-add of V
    edge_aggregate_kernel<<<nblk_edge, 256, 0, stream>>>(V, ei, alpha, segsum, h, E);
}